// difficultyWeightedLoss_25486335934801
// MI455X (gfx1250) — compile-verified
//
#include <hip/hip_runtime.h>

typedef float v2f __attribute__((ext_vector_type(2)));
typedef float v4f __attribute__((ext_vector_type(4)));
typedef float v8f __attribute__((ext_vector_type(8)));
typedef int   v4i __attribute__((ext_vector_type(4)));

#define TPB    256          // 8 waves (wave32)
#define CHUNKS 4            // vector chunks per thread
#define SPT    (CHUNKS * 4) // samples per thread = 16
#define SPB    (TPB * SPT)  // samples per block = 4096

// ce = softplus(l_other - l_target); weight 1.4 for target==0, 1.0 for target==1
__device__ __forceinline__ float sample_loss(float l0, float l1, int t) {
    float z  = (t != 0) ? (l0 - l1) : (l1 - l0);
    float ce = fmaxf(z, 0.0f) + __logf(1.0f + __expf(-fabsf(z)));
    return ((t != 0) ? 1.0f : 1.4f) * ce;
}

// 32-lane sum via V_WMMA_F32_16X16X4_F32 (requires EXEC == all ones).
// A layout (32-bit 16x4): VGPR0 lanes0-15 = A[m,0], lanes16-31 = A[m,2];
// VGPR1 = K=1/K=3 (zeroed).  B = ones(4x16) -> D[m,n] = s[m] + s[m+16].
// D VGPR r: lanes0-15 hold M=r, lanes16-31 hold M=r+8 -> per-lane sum of the
// 8 D regs gives rows 0-7 (low half) / rows 8-15 (high half); one xor-16
// shuffle finishes the reduction.
__device__ __forceinline__ float wave_reduce(float s) {
    v2f a; a.x = s;    a.y = 0.0f;
    v2f b; b.x = 1.0f; b.y = 1.0f;
    v8f c = {};
    v8f d = __builtin_amdgcn_wmma_f32_16x16x4_f32(
        /*neg_a=*/false, a, /*neg_b=*/false, b,
        /*c_mod=*/(short)0, c, /*reuse_a=*/false, /*reuse_b=*/false);
    float t = ((d[0] + d[1]) + (d[2] + d[3])) + ((d[4] + d[5]) + (d[6] + d[7]));
    t += __shfl_xor(t, 16, 32);
    return t;
}

__device__ __forceinline__ float block_reduce(float s, float* lds) {
    float w = wave_reduce(s);
    int wave = threadIdx.x >> 5;
    if ((threadIdx.x & 31) == 0) lds[wave] = w;
    __syncthreads();
    float tot = 0.0f;
#pragma unroll
    for (int i = 0; i < TPB / 32; ++i) tot += lds[i];
    return tot;
}

__global__ void __launch_bounds__(TPB)
dwl_partial(const float* __restrict__ logits, const int* __restrict__ targets,
            float* __restrict__ partials, int n) {
    __shared__ float lds[TPB / 32];
    const int blockStart = (int)blockIdx.x * SPB;
    float acc = 0.0f;
#pragma unroll
    for (int j = 0; j < CHUNKS; ++j) {
        int base = blockStart + j * (TPB * 4) + (int)threadIdx.x * 4;
        if (base + 3 < n) {
            // 4 samples: 2x B128 NT loads of logits, 1x B128 NT load of targets
            const v4f* lp = (const v4f*)(logits + 2 * (size_t)base);
            v4f p  = __builtin_nontemporal_load(lp);
            v4f q  = __builtin_nontemporal_load(lp + 1);
            v4i tv = __builtin_nontemporal_load((const v4i*)(targets + base));
            acc += sample_loss(p.x, p.y, tv.x);
            acc += sample_loss(p.z, p.w, tv.y);
            acc += sample_loss(q.x, q.y, tv.z);
            acc += sample_loss(q.z, q.w, tv.w);
        } else {
            for (int i = base; i < n && i < base + 4; ++i)
                acc += sample_loss(logits[2 * (size_t)i],
                                   logits[2 * (size_t)i + 1], targets[i]);
        }
    }
    // All lanes re-converged here: EXEC all-ones for the WMMA reduction.
    float tot = block_reduce(acc, lds);
    if (threadIdx.x == 0) partials[blockIdx.x] = tot;
}

__global__ void __launch_bounds__(TPB)
dwl_final(const float* __restrict__ partials, float* __restrict__ out,
          int numPartials, float invN) {
    __shared__ float lds[TPB / 32];
    float acc = 0.0f;
    for (int i = (int)threadIdx.x; i < numPartials; i += TPB)
        acc += partials[i];
    float tot = block_reduce(acc, lds);
    if (threadIdx.x == 0) out[0] = tot * invN;
}

extern "C" void kernel_launch(void* const* d_in, const int* in_sizes, int n_in,
                              void* d_out, int out_size, void* d_ws, size_t ws_size,
                              hipStream_t stream) {
    const float* logits  = (const float*)d_in[0];
    const int*   targets = (const int*)d_in[1];
    // d_in[2] (text_keys) is unused by the reference.
    (void)n_in; (void)out_size; (void)ws_size;

    const int n = in_sizes[1];                 // N samples
    float* partials = (float*)d_ws;            // one float per block
    const int blocks = (n + SPB - 1) / SPB;    // 4096 for N = 16.7M

    dwl_partial<<<blocks, TPB, 0, stream>>>(logits, targets, partials, n);
    dwl_final<<<1, TPB, 0, stream>>>(partials, (float*)d_out, blocks,
                                     1.0f / (float)n);
}